// EdgeCentricNetwork_25039659335710
// MI455X (gfx1250) — compile-verified
//
#include <hip/hip_runtime.h>

// EdgeCentricNetwork fused kernel for MI455X (gfx1250, wave32).
//
// delta[b,j] = sum_{i,e} relu( (X W_src)[b,i,e] + (X W_dst)[b,j,e] + be[e] ) * Wr[i*E+e] + br
//
// Stage 1: f32 WMMA projection (exact f32 math, V_WMMA_F32_16X16X4_F32), S/T -> d_ws.
// Stage 2: fused relu-weighted contraction, edges tensor never materialized.

typedef float v2f __attribute__((ext_vector_type(2)));
typedef float v8f __attribute__((ext_vector_type(8)));

#define BB 8
#define NN 512
#define DD 64
#define EE 32
#define ROWS (BB * NN)   // 4096
#define JPW 4            // outputs per wave in stage 2

// ---------------- Stage 1: S = X @ We[:D], T = X @ We[D:] + be ----------------
// One wave per 16-row tile of X; computes 4 output tiles (S cols 0-15, S cols
// 16-31, T cols 0-15, T cols 16-31) chaining 16 WMMA steps over K = 64.
__global__ __launch_bounds__(256) void proj_wmma_kernel(
    const float* __restrict__ x,   // [ROWS][D]
    const float* __restrict__ We,  // [2*D][E]
    const float* __restrict__ be,  // [E]
    float* __restrict__ S,         // [ROWS][E]
    float* __restrict__ T)         // [ROWS][E]
{
    const int tid  = blockIdx.x * blockDim.x + threadIdx.x;
    const int wave = tid >> 5;
    const int lane = tid & 31;
    const int hi   = lane >> 4;    // 0: lanes 0-15, 1: lanes 16-31
    const int mn   = lane & 15;    // A row / B,C,D column within tile
    const int row0 = wave * 16;

    v8f accS0 = {}; v8f accS1 = {}; v8f accT0 = {}; v8f accT1 = {};

    const float* xrow = x + (size_t)(row0 + mn) * DD;

    #pragma unroll
    for (int kb = 0; kb < DD; kb += 4) {
        const int k = kb + 2 * hi;
        // A fragment 16x4 f32: elem v = A[m][kb + 2*hi + v]
        v2f a;
        a.x = xrow[k];
        a.y = xrow[k + 1];
        // B fragments 4x16 f32: elem v = B[kb + 2*hi + v][n]
        v2f bs0, bs1, bd0, bd1;
        bs0.x = We[(k    ) * EE + mn];      bs0.y = We[(k + 1) * EE + mn];
        bs1.x = We[(k    ) * EE + 16 + mn]; bs1.y = We[(k + 1) * EE + 16 + mn];
        bd0.x = We[(DD + k    ) * EE + mn];      bd0.y = We[(DD + k + 1) * EE + mn];
        bd1.x = We[(DD + k    ) * EE + 16 + mn]; bd1.y = We[(DD + k + 1) * EE + 16 + mn];

        accS0 = __builtin_amdgcn_wmma_f32_16x16x4_f32(false, a, false, bs0, (short)0, accS0, false, false);
        accS1 = __builtin_amdgcn_wmma_f32_16x16x4_f32(false, a, false, bs1, (short)0, accS1, false, false);
        accT0 = __builtin_amdgcn_wmma_f32_16x16x4_f32(false, a, false, bd0, (short)0, accT0, false, false);
        accT1 = __builtin_amdgcn_wmma_f32_16x16x4_f32(false, a, false, bd1, (short)0, accT1, false, false);
    }

    const float be0 = be[mn];
    const float be1 = be[16 + mn];

    // C/D layout: elem v sits at (M = v + 8*hi, N = mn)
    #pragma unroll
    for (int v = 0; v < 8; ++v) {
        const int r = row0 + v + 8 * hi;
        S[(size_t)r * EE + mn]      = accS0[v];
        S[(size_t)r * EE + 16 + mn] = accS1[v];
        T[(size_t)r * EE + mn]      = accT0[v] + be0;
        T[(size_t)r * EE + 16 + mn] = accT1[v] + be1;
    }
}

// ---------------- Stage 2: fused relu contraction ----------------
// One wave per JPW consecutive outputs j within batch b. Lane = e channel.
// Streams S[b,i,:] and Wr[i,:] (hot in WGP$/L2), reuses each row across JPW
// accumulators. Edges tensor (268 MB) never touches memory.
__global__ __launch_bounds__(256) void edge_reduce_kernel(
    const float* __restrict__ S,   // [ROWS][E]
    const float* __restrict__ T,   // [ROWS][E]  (already includes +be)
    const float* __restrict__ Wr,  // [N*E]
    const float* __restrict__ br,  // [1]
    float* __restrict__ out)       // [B*N]
{
    const int tid  = blockIdx.x * blockDim.x + threadIdx.x;
    const int wave = tid >> 5;
    const int lane = tid & 31;              // e index
    const int b    = wave / (NN / JPW);
    const int j0   = (wave % (NN / JPW)) * JPW;

    float t[JPW], acc[JPW];
    #pragma unroll
    for (int jj = 0; jj < JPW; ++jj) {
        t[jj]   = T[((size_t)(b * NN + j0 + jj)) * EE + lane];
        acc[jj] = 0.0f;
    }

    const float* Sb = S + (size_t)b * NN * EE;

    #pragma unroll 4
    for (int i = 0; i < NN; ++i) {
        const float s = Sb[(size_t)i * EE + lane];
        const float w = Wr[(size_t)i * EE + lane];
        #pragma unroll
        for (int jj = 0; jj < JPW; ++jj) {
            const float e = fmaxf(s + t[jj], 0.0f);   // relu(src + dst + be)
            acc[jj] = fmaf(e, w, acc[jj]);
        }
    }

    const float brv = br[0];
    #pragma unroll
    for (int jj = 0; jj < JPW; ++jj) {
        float v = acc[jj];
        #pragma unroll
        for (int off = 16; off > 0; off >>= 1)
            v += __shfl_xor(v, off, 32);              // wave32 tree reduction
        if (lane == 0)
            out[b * NN + j0 + jj] = v + brv;
    }
}

extern "C" void kernel_launch(void* const* d_in, const int* in_sizes, int n_in,
                              void* d_out, int out_size, void* d_ws, size_t ws_size,
                              hipStream_t stream) {
    const float* x  = (const float*)d_in[0];  // [B,N,D]
    const float* We = (const float*)d_in[1];  // [2D,E]
    const float* be = (const float*)d_in[2];  // [E]
    const float* Wr = (const float*)d_in[3];  // [N*E,1]
    const float* br = (const float*)d_in[4];  // [1]
    float* out = (float*)d_out;               // [B,N,1] = 4096 floats

    float* S = (float*)d_ws;                  // ROWS*E floats = 512 KB
    float* T = S + (size_t)ROWS * EE;         // another 512 KB

    // Stage 1: 4096/16 = 256 waves -> 8192 threads -> 32 blocks of 256.
    proj_wmma_kernel<<<(ROWS / 16) * 32 / 256, 256, 0, stream>>>(x, We, be, S, T);

    // Stage 2: 8*512/JPW = 1024 waves -> 32768 threads -> 128 blocks of 256.
    edge_reduce_kernel<<<(BB * NN / JPW) * 32 / 256, 256, 0, stream>>>(S, T, Wr, br, out);
}